// GNN_54099408060932
// MI455X (gfx1250) — compile-verified
//
#include <hip/hip_runtime.h>
#include <hip/hip_bf16.h>
#include <math.h>

#define N_NODES 100000
#define N_EDGES 600000
#define DIM     128
#define NLAYERS 4
#define NGRAPH  512
#define NCLASS  40
#define LN_EPSF 1e-5f

typedef __attribute__((ext_vector_type(2))) float v2f;
typedef __attribute__((ext_vector_type(8))) float v8f;

// t[n,d] = 2*h[n,d]   (covers the self-loop contribution + the extra "+h")
__global__ void gin_init_t(const float* __restrict__ h, float* __restrict__ t, int n4) {
    int i = blockIdx.x * blockDim.x + threadIdx.x;
    if (i < n4) {
        float4 v = ((const float4*)h)[i];
        v.x *= 2.f; v.y *= 2.f; v.z *= 2.f; v.w *= 2.f;
        ((float4*)t)[i] = v;
    }
}

// t[dst[e], :] += h[src[e], :]  — one wave (32 lanes x 4 floats) per edge
__global__ void gin_scatter(const float* __restrict__ h, const int* __restrict__ src,
                            const int* __restrict__ dst, float* __restrict__ t) {
    int tid = blockIdx.x * blockDim.x + threadIdx.x;
    int e = tid >> 5;
    int lane = tid & 31;
    if (e < N_EDGES) {
        int s = src[e], d = dst[e];
        const float4 v = *(const float4*)(h + (long long)s * DIM + lane * 4);
        float* p = t + (long long)d * DIM + lane * 4;
        atomicAdd(p + 0, v.x);
        atomicAdd(p + 1, v.y);
        atomicAdd(p + 2, v.z);
        atomicAdd(p + 3, v.w);
    }
}

// Fused: Hout = LayerNorm(relu(T @ W + bias)) * gamma + beta
// block = 256 threads = 8 waves; block handles 16 rows x 128 cols; each wave
// computes one 16x16 tile with V_WMMA_F32_16X16X4_F32 (K-loop of 32).
__global__ __launch_bounds__(256)
void gin_gemm_ln(const float* __restrict__ T, const float* __restrict__ W,
                 const float* __restrict__ bias, const float* __restrict__ gamma,
                 const float* __restrict__ beta, float* __restrict__ Hout) {
    __shared__ float red_s[16][8];
    __shared__ float red_q[16][8];
    __shared__ float mu[16];
    __shared__ float rstd[16];

    const int wave = threadIdx.x >> 5;
    const int lane = threadIdx.x & 31;
    const int half = lane >> 4;     // which 16-lane half of the wave
    const int l16  = lane & 15;
    const int row0 = blockIdx.x * 16;
    const int rowA = row0 + l16;            // A-matrix row held by this lane
    const int colB = wave * 16 + l16;       // B/C/D column held by this lane

    // A f32 16x4 layout: v0 = K = k0+2*half, v1 = K = k0+2*half+1 (rows on lanes)
    // B f32 4x16 layout: v0 = K = k0+2*half, v1 = K = k0+2*half+1 (cols on lanes)
    const float* arow = T + (long long)rowA * DIM + 2 * half;

    v8f acc = {};
#pragma unroll
    for (int k0 = 0; k0 < DIM; k0 += 4) {
        v2f a, b;
        a.x = arow[k0];
        a.y = arow[k0 + 1];
        const int kb = k0 + 2 * half;
        b.x = W[kb * DIM + colB];
        b.y = W[(kb + 1) * DIM + colB];
        acc = __builtin_amdgcn_wmma_f32_16x16x4_f32(
                false, a, false, b, (short)0, acc, false, false);
    }

    const float bv = bias[colB];
    const float gv = gamma[colB];
    const float bt = beta[colB];

    // bias + relu, kept in registers; C layout: VGPR r -> row (r + 8*half)
    float vals[8];
#pragma unroll
    for (int r = 0; r < 8; ++r) {
        float v = acc[r] + bv;
        vals[r] = v > 0.f ? v : 0.f;
    }

    // per-row partial sum / sumsq over this wave's 16 columns
#pragma unroll
    for (int r = 0; r < 8; ++r) {
        float s = vals[r];
        float q = vals[r] * vals[r];
#pragma unroll
        for (int m = 1; m < 16; m <<= 1) {
            s += __shfl_xor(s, m, 32);
            q += __shfl_xor(q, m, 32);
        }
        if (l16 == 0) {
            const int rt = r + 8 * half;
            red_s[rt][wave] = s;
            red_q[rt][wave] = q;
        }
    }
    __syncthreads();

    if (threadIdx.x < 16) {
        float s = 0.f, q = 0.f;
#pragma unroll
        for (int w = 0; w < 8; ++w) { s += red_s[threadIdx.x][w]; q += red_q[threadIdx.x][w]; }
        const float m   = s * (1.0f / DIM);
        const float var = q * (1.0f / DIM) - m * m;
        mu[threadIdx.x]   = m;
        rstd[threadIdx.x] = rsqrtf(var + LN_EPSF);
    }
    __syncthreads();

#pragma unroll
    for (int r = 0; r < 8; ++r) {
        const int mrow = r + 8 * half;
        const float o = (vals[r] - mu[mrow]) * rstd[mrow] * gv + bt;
        Hout[(long long)(row0 + mrow) * DIM + colB] = o;
    }
}

__global__ void gin_zero(float* __restrict__ p, int n) {
    int i = blockIdx.x * blockDim.x + threadIdx.x;
    if (i < n) p[i] = 0.f;
}

// segment-sum pooling: one wave per node
__global__ void gin_pool(const float* __restrict__ h, const int* __restrict__ batch,
                         float* __restrict__ psum, float* __restrict__ cnt) {
    int tid = blockIdx.x * blockDim.x + threadIdx.x;
    int n = tid >> 5;
    int lane = tid & 31;
    if (n < N_NODES) {
        int b = batch[n];
        const float4 v = *(const float4*)(h + (long long)n * DIM + lane * 4);
        float* p = psum + (long long)b * DIM + lane * 4;
        atomicAdd(p + 0, v.x);
        atomicAdd(p + 1, v.y);
        atomicAdd(p + 2, v.z);
        atomicAdd(p + 3, v.w);
        if (lane == 0) atomicAdd(cnt + b, 1.0f);
    }
}

// logits[g,c] = b_out[c] + (psum[g,:]/max(cnt,1)) @ W_out[:,c]
__global__ void gin_outproj(const float* __restrict__ psum, const float* __restrict__ cnt,
                            const float* __restrict__ Wout, const float* __restrict__ bout,
                            float* __restrict__ logits) {
    int idx = blockIdx.x * blockDim.x + threadIdx.x;
    if (idx < NGRAPH * NCLASS) {
        int g = idx / NCLASS;
        int c = idx - g * NCLASS;
        float inv = 1.0f / fmaxf(cnt[g], 1.0f);
        float acc = bout[c];
        const float* pg = psum + g * DIM;
#pragma unroll 4
        for (int k = 0; k < DIM; ++k) acc += pg[k] * inv * Wout[k * NCLASS + c];
        logits[idx] = acc;
    }
}

__global__ void gin_logsoftmax(const float* __restrict__ logits, float* __restrict__ out) {
    __shared__ float sh[64];
    const int g = blockIdx.x;
    const int t = threadIdx.x;
    float v = (t < NCLASS) ? logits[g * NCLASS + t] : -3.4e38f;
    sh[t] = v;
    __syncthreads();
    for (int s = 32; s > 0; s >>= 1) {
        if (t < s) sh[t] = fmaxf(sh[t], sh[t + s]);
        __syncthreads();
    }
    const float mx = sh[0];
    __syncthreads();
    sh[t] = (t < NCLASS) ? expf(v - mx) : 0.f;
    __syncthreads();
    for (int s = 32; s > 0; s >>= 1) {
        if (t < s) sh[t] += sh[t + s];
        __syncthreads();
    }
    const float lse = logf(sh[0]);
    if (t < NCLASS) out[g * NCLASS + t] = v - mx - lse;
}

extern "C" void kernel_launch(void* const* d_in, const int* in_sizes, int n_in,
                              void* d_out, int out_size, void* d_ws, size_t ws_size,
                              hipStream_t stream) {
    const float* x      = (const float*)d_in[0];
    const int*   ei     = (const int*)d_in[1];     // [2,E]: row0=src, row1=dst
    const int*   batch  = (const int*)d_in[2];
    const float* Ws     = (const float*)d_in[3];   // [L,D,D]
    const float* bs     = (const float*)d_in[4];   // [L,D]
    const float* gammas = (const float*)d_in[5];   // [L,D]
    const float* betas  = (const float*)d_in[6];   // [L,D]
    const float* Wout   = (const float*)d_in[7];   // [D,C]
    const float* bout   = (const float*)d_in[8];   // [C]
    float* out = (float*)d_out;

    const int* src = ei;
    const int* dst = ei + N_EDGES;

    float* ws = (float*)d_ws;
    const size_t nd = (size_t)N_NODES * DIM;
    float* hA     = ws;
    float* hB     = ws + nd;
    float* tbuf   = ws + 2 * nd;
    float* psum   = ws + 3 * nd;
    float* cnt    = psum + (size_t)NGRAPH * DIM;
    float* logits = cnt + NGRAPH;

    const int n4 = (N_NODES * DIM) / 4;
    const int gInit    = (n4 + 255) / 256;
    const int gScatter = (N_EDGES * 32 + 255) / 256;
    const int gGemm    = N_NODES / 16;   // 6250, exact
    const int gPool    = (N_NODES * 32 + 255) / 256;

    const float* hin = x;
    for (int i = 0; i < NLAYERS; ++i) {
        float* hout = (i & 1) ? hB : hA;
        gin_init_t<<<gInit, 256, 0, stream>>>(hin, tbuf, n4);
        gin_scatter<<<gScatter, 256, 0, stream>>>(hin, src, dst, tbuf);
        gin_gemm_ln<<<gGemm, 256, 0, stream>>>(tbuf, Ws + (size_t)i * DIM * DIM,
                                               bs + i * DIM, gammas + i * DIM,
                                               betas + i * DIM, hout);
        hin = hout;
    }

    const int nPool = NGRAPH * DIM + NGRAPH;
    gin_zero<<<(nPool + 255) / 256, 256, 0, stream>>>(psum, nPool);
    gin_pool<<<gPool, 256, 0, stream>>>(hin, batch, psum, cnt);
    gin_outproj<<<(NGRAPH * NCLASS + 255) / 256, 256, 0, stream>>>(psum, cnt, Wout, bout, logits);
    gin_logsoftmax<<<NGRAPH, 64, 0, stream>>>(logits, out);
}